// channel_clustering_hpro_loss_79757542686982
// MI455X (gfx1250) — compile-verified
//
#include <hip/hip_runtime.h>
#include <stdint.h>

#ifndef __has_builtin
#define __has_builtin(x) 0
#endif

// CDNA5 async global->LDS path (gfx1250): guard so host pass / older toolchains
// fall back to plain vector loads and the file always compiles.
#if defined(__gfx1250__) && __has_builtin(__builtin_amdgcn_global_load_async_to_lds_b128)
#define USE_ASYNC_LDS 1
#else
#define USE_ASYNC_LDS 0
#endif

#define AS1 __attribute__((address_space(1)))
#define AS3 __attribute__((address_space(3)))
typedef int v4i __attribute__((vector_size(16)));   // matches builtin param pointee

#if __has_builtin(__builtin_amdgcn_s_wait_asynccnt)
#define WAIT_ASYNC_0() __builtin_amdgcn_s_wait_asynccnt(0)
#else
#define WAIT_ASYNC_0() asm volatile("s_wait_asynccnt 0" ::: "memory")
#endif

#define BB   8
#define RR   32
#define WWH  48
#define HH   48
#define DDp  48
#define HD   2304          // H*D
#define WHD  110592        // W*H*D (voxels per (b,r))
#define NBR  256           // B*R
#define BLK2 432           // blocks per b in pass 2 (110592/256)
#define NPART 3456         // 8*432
#define TOTF 28311552.0f   // B*R*W*H*D

// ---------------- Pass 1: per-(b,r) argmax + sum (one block per (b,r)) -------
__global__ __launch_bounds__(256) void k_argmax_sum(const float* __restrict__ feat,
                                                    int* __restrict__ outIdx,
                                                    float* __restrict__ outSum) {
  const int tid = threadIdx.x;
  const int br  = blockIdx.x;
  const float4* base = (const float4*)(feat + (size_t)br * WHD);  // 16B aligned
  float bestv = -3.402823e38f;
  int   besti = 0;
  float sum   = 0.f;
  // 110592 floats = 27648 float4 = 108 iters * 256 threads, coalesced
  for (int i = 0; i < 108; ++i) {
    const int c = i * 256 + tid;
    const float4 v = base[c];
    const int e = c * 4;
    if (v.x > bestv) { bestv = v.x; besti = e;     }
    if (v.y > bestv) { bestv = v.y; besti = e + 1; }
    if (v.z > bestv) { bestv = v.z; besti = e + 2; }
    if (v.w > bestv) { bestv = v.w; besti = e + 3; }
    sum += v.x + v.y + v.z + v.w;
  }
  __shared__ float sv[256]; __shared__ int si[256]; __shared__ float ss[256];
  sv[tid] = bestv; si[tid] = besti; ss[tid] = sum;
  __syncthreads();
  for (int s = 128; s > 0; s >>= 1) {        // deterministic tree reduce
    if (tid < s) {
      const float ov = sv[tid + s]; const int oi = si[tid + s];
      if (ov > sv[tid] || (ov == sv[tid] && oi < si[tid])) { sv[tid] = ov; si[tid] = oi; }
      ss[tid] += ss[tid + s];
    }
    __syncthreads();
  }
  if (tid == 0) { outIdx[br] = si[0]; outSum[br] = ss[0]; }
}

// ---------------- Pass 1.5: mgr, dz2 total, mx/my (single block) -------------
__global__ __launch_bounds__(256) void k_prep(const int* __restrict__ idxArr,
                                              const float* __restrict__ sumArr,
                                              float* __restrict__ wsMx,
                                              float* __restrict__ wsMy,
                                              float* __restrict__ wsMgr,
                                              float* __restrict__ wsDz) {
  const int t = threadIdx.x;                 // t == (b*R + r)
  const int idx = idxArr[t];
  const int mx = idx / HD;
  const int my = (idx / DDp) % HH;
  const int mz = idx % DDp;
  wsMx[t] = (float)mx;
  wsMy[t] = (float)my;
  float dz = 0.f;
  for (int k = 0; k < DDp; ++k) { const float d = (float)(mz - k); dz += d * d; }
  dz *= (float)(WWH * HH);                   // broadcast of dz2 over (W,H)
  __shared__ float sa[256], sb[256];
  sa[t] = sumArr[t]; sb[t] = dz;
  __syncthreads();
  for (int s = 128; s > 0; s >>= 1) {
    if (t < s) { sa[t] += sa[t + s]; sb[t] += sb[t + s]; }
    __syncthreads();
  }
  if (t == 0) { wsMgr[0] = sa[0] / TOTF; wsDz[0] = sb[0]; }
}

// ---------------- Pass 2: per-voxel top-2 over R + weighted sums -------------
__global__ __launch_bounds__(256) void k_main(const float* __restrict__ feat,
                                              const float* __restrict__ wsMx,
                                              const float* __restrict__ wsMy,
                                              const float* __restrict__ wsMgr,
                                              float* __restrict__ part) {
  __shared__ __align__(16) float lbuf[RR * 256];   // 32 KB staged tile
  __shared__ float smx[RR], smy[RR];
  __shared__ float rd[256], rv[256];
  const int tid      = threadIdx.x;
  const int b        = blockIdx.x / BLK2;
  const int vox_base = (blockIdx.x % BLK2) * 256;

  if (tid < RR) {
    smx[tid] = wsMx[b * RR + tid];
    smy[tid] = wsMy[b * RR + tid];
  }

  const float* featb = feat + (size_t)b * RR * WHD + vox_base;
  // Stage 32 channels x 256 voxels: 2048 16B chunks, 8 per thread, coalesced.
#pragma unroll
  for (int it = 0; it < 8; ++it) {
    const int c   = it * 256 + tid;     // chunk id: floats [4c, 4c+4)
    const int r   = c >> 6;             // 64 chunks per channel slice
    const int pos = (c & 63) << 2;
    const float* gp = featb + (size_t)r * WHD + pos;
    float* lp = &lbuf[c << 2];
#if USE_ASYNC_LDS
    __builtin_amdgcn_global_load_async_to_lds_b128((AS1 v4i*)gp, (AS3 v4i*)lp, 0, 0);
#else
    *(float4*)lp = *(const float4*)gp;
#endif
  }
#if USE_ASYNC_LDS
  WAIT_ASYNC_0();                       // each wave drains its own ASYNCcnt
#endif
  __syncthreads();                      // make all waves' LDS writes visible

  const float mgr   = wsMgr[0];
  const int   voxel = vox_base + tid;
  const float wf = (float)(voxel / HD);
  const float hf = (float)((voxel / DDp) % HH);

  float m1 = -3.402823e38f, m2 = -3.402823e38f, sf2 = 0.f, dis = 0.f;
#pragma unroll
  for (int r = 0; r < RR; ++r) {
    const float f  = lbuf[r * 256 + tid];    // conflict-free: bank = tid%64
    const float f2 = f * f;
    const float cx = smx[r] - wf;
    const float cy = smy[r] - hf;
    dis = fmaf(f2, fmaf(cx, cx, cy * cy), dis);
    sf2 += f2;
    if (f > m1)      { m2 = m1; m1 = f; }
    else if (f > m2) { m2 = f; }
  }
  // Σ_r f²(sel−mgr)² with sel=m1 except argmax channel (where f=m1) uses m2:
  const float a1 = m1 - mgr, a2 = m2 - mgr;
  const float p  = m1 * m1;                  // f² at the argmax channel
  const float dv = a1 * a1 * (sf2 - p) + a2 * a2 * p;

  rd[tid] = dis; rv[tid] = dv;
  __syncthreads();
  for (int s = 128; s > 0; s >>= 1) {
    if (tid < s) { rd[tid] += rd[tid + s]; rv[tid] += rv[tid + s]; }
    __syncthreads();
  }
  if (tid == 0) { part[2 * blockIdx.x] = rd[0]; part[2 * blockIdx.x + 1] = rv[0]; }
}

// ---------------- Final deterministic reduction ------------------------------
__global__ __launch_bounds__(256) void k_final(const float* __restrict__ part,
                                               const float* __restrict__ wsDz,
                                               float* __restrict__ out) {
  const int t = threadIdx.x;
  float a = 0.f, v = 0.f;
  for (int i = t; i < NPART; i += 256) { a += part[2 * i]; v += part[2 * i + 1]; }
  __shared__ float sa[256], sv[256];
  sa[t] = a; sv[t] = v;
  __syncthreads();
  for (int s = 128; s > 0; s >>= 1) {
    if (t < s) { sa[t] += sa[t + s]; sv[t] += sv[t + s]; }
    __syncthreads();
  }
  if (t == 0) {
    out[0] = (sa[0] + wsDz[0]) / TOTF;   // dis_loss / sum_num
    out[1] = sv[0] / TOTF;               // div_loss / sum_num
  }
}

extern "C" void kernel_launch(void* const* d_in, const int* in_sizes, int n_in,
                              void* d_out, int out_size, void* d_ws, size_t ws_size,
                              hipStream_t stream) {
  (void)in_sizes; (void)n_in; (void)out_size; (void)ws_size;
  // d_in[0] backbone_feature, d_in[1] grouping_result: dead inputs.
  const float* feat = (const float*)d_in[2];
  float* ws    = (float*)d_ws;
  int*   wsIdx = (int*)ws;          // [256]
  float* wsSum = ws + 256;          // [256]
  float* wsMx  = ws + 512;          // [256]
  float* wsMy  = ws + 768;          // [256]
  float* wsMgr = ws + 1024;         // [1]
  float* wsDz  = ws + 1025;         // [1]
  float* wsPart= ws + 1032;         // [3456*2]
  float* out   = (float*)d_out;

  k_argmax_sum<<<NBR,   256, 0, stream>>>(feat, wsIdx, wsSum);
  k_prep      <<<1,     256, 0, stream>>>(wsIdx, wsSum, wsMx, wsMy, wsMgr, wsDz);
  k_main      <<<NPART, 256, 0, stream>>>(feat, wsMx, wsMy, wsMgr, wsPart);
  k_final     <<<1,     256, 0, stream>>>(wsPart, wsDz, out);
}